// ConditionalClipSegVoxelDiffusionModel_68762426409353
// MI455X (gfx1250) — compile-verified
//
#include <hip/hip_runtime.h>
#include <stdint.h>

// ---------------- problem constants (match reference) ----------------
#define BATCHES 8
#define NPTS    500000
#define NPOW2   524288          // 2^19 >= NPTS (point-key sort size)
#define IDXBITS 19
#define IDXMASK ((1u << IDXBITS) - 1)
#define NVOX    64000           // 40*40*40
#define NBINS   64001           // + sentinel bucket
#define VSORT   65536           // 2^16 >= NVOX (voxel ranking sort size)
#define VTOP    4096
#define PPV     64
#define CSTRIDE 64004           // padded NBINS, 16B aligned

// LDS bitonic tile: 4096 u64 keys = 32 KB (WGP has 320 KB LDS)
#define TILE    4096
#define TTHR    1024

// ---------------- workspace layout (bytes) ----------------
#define WS_BASE   0                                            // 8 x u32  (max|pc| bits)
#define WS_COUNTS 64                                           // 8 x CSTRIDE x u32
#define WS_STARTS (WS_COUNTS + (size_t)BATCHES * CSTRIDE * 4)  // 8 x CSTRIDE x u32
#define WS_PKEYS  (WS_STARTS + (size_t)BATCHES * CSTRIDE * 4)  // 8 x NPOW2 x u64
#define WS_RKEYS  (WS_PKEYS + (size_t)BATCHES * NPOW2 * 8)     // 8 x VSORT x u64

// ---------------- CDNA5 async Global<->LDS path (guarded) ----------------
#if defined(__HIP_DEVICE_COMPILE__) && __has_builtin(__builtin_amdgcn_global_load_async_to_lds_b128) && \
    __has_builtin(__builtin_amdgcn_global_store_async_from_lds_b128)
#define HAVE_ASYNC_LDS 1
#else
#define HAVE_ASYNC_LDS 0
#endif

#if defined(__HIP_DEVICE_COMPILE__)
__device__ __forceinline__ void wait_asynccnt0() {
#if __has_builtin(__builtin_amdgcn_s_wait_asynccnt)
  __builtin_amdgcn_s_wait_asynccnt(0);
#else
  asm volatile("s_wait_asynccnt 0x0" ::: "memory");
#endif
}
#endif

#if HAVE_ASYNC_LDS
// b128 payload type: int __attribute__((vector_size(16))) (b64 was vector_size(8), diagnosed)
typedef int v4i_t __attribute__((vector_size(4 * sizeof(int))));
typedef __attribute__((address_space(1))) v4i_t gas_v4i;   // HIP prints AS1 as "__device__"
typedef __attribute__((address_space(3))) v4i_t las_v4i;   // HIP prints AS3 as "__shared__"
// generic LDS address: low 32 bits are the LDS byte offset (aperture truncation)
__device__ __forceinline__ las_v4i* to_lds(unsigned long long* p) {
  return (las_v4i*)(size_t)(unsigned)(size_t)p;
}
__device__ __forceinline__ gas_v4i* to_glob(unsigned long long* p) {
  return (gas_v4i*)(size_t)p;
}
#endif

// ---------------- kernels ----------------

__global__ void k_zero(unsigned* p, int n) {
  int i = blockIdx.x * blockDim.x + threadIdx.x;
  if (i < n) p[i] = 0u;
}

// base[b] = max |pc[b,:,:]|  (positive-float max == u32 max on bit patterns)
__global__ void k_reduce_base(const float* __restrict__ pc, unsigned* baseBits) {
  const int b = blockIdx.y;
  const float* src = pc + (size_t)b * NPTS * 3;
  const int n = NPTS * 3;
  float m = 0.0f;
  for (int i = blockIdx.x * blockDim.x + threadIdx.x; i < n;
       i += gridDim.x * blockDim.x) {
    __builtin_prefetch(src + i + 16384, 0, 1);   // gfx1250 global_prefetch_b8
    m = fmaxf(m, fabsf(src[i]));
  }
  __shared__ float red[256];
  red[threadIdx.x] = m;
  __syncthreads();
  for (int s = 128; s > 0; s >>= 1) {
    if (threadIdx.x < s) red[threadIdx.x] = fmaxf(red[threadIdx.x], red[threadIdx.x + s]);
    __syncthreads();
  }
  if (threadIdx.x == 0) atomicMax(&baseBits[b], __float_as_uint(red[0]));
}

// per-point: voxel id, histogram, and stable sort key (vid<<19 | point_idx)
__global__ void k_bin(const float* __restrict__ pc, const unsigned* __restrict__ baseBits,
                      unsigned* counts, unsigned long long* pkeys) {
  const int b = blockIdx.y;
  const int i = blockIdx.x * blockDim.x + threadIdx.x;
  if (i >= NPOW2) return;
  unsigned long long* pk = pkeys + (size_t)b * NPOW2;
  if (i >= NPTS) { pk[i] = ~0ull; return; }           // padding sorts last
  const float base = __uint_as_float(baseBits[b]);
  const float* s = pc + ((size_t)b * NPTS + i) * 3;
  __builtin_prefetch(s + 3 * 2048, 0, 1);
  const float x = s[0] / base, y = s[1] / base, z = s[2] / base;
  const int ix = (int)floorf((x + 1.0f) / 0.05f);
  const int iy = (int)floorf((y + 1.0f) / 0.05f);
  const int iz = (int)floorf((z + 1.0f) / 0.05f);
  int vid;
  if (ix >= 0 && ix < 40 && iy >= 0 && iy < 40 && iz >= 0 && iz < 40)
    vid = ix * 1600 + iy * 40 + iz;
  else
    vid = NVOX;
  atomicAdd(&counts[(size_t)b * CSTRIDE + vid], 1u);
  pk[i] = ((unsigned long long)(unsigned)vid << IDXBITS) | (unsigned)i;
}

// exclusive prefix sum of counts[NBINS] -> starts, one block (1024 thr) per batch
__global__ void k_scan(const unsigned* __restrict__ counts, unsigned* starts) {
  const int b = blockIdx.x;
  const unsigned* c = counts + (size_t)b * CSTRIDE;
  unsigned* st = starts + (size_t)b * CSTRIDE;
  __shared__ unsigned buf[1024];
  unsigned carry = 0;
  for (int chunk = 0; chunk < 63; ++chunk) {          // 63*1024 >= NBINS
    const int idx = chunk * 1024 + threadIdx.x;
    const unsigned v = (idx < NBINS) ? c[idx] : 0u;
    buf[threadIdx.x] = v;
    __syncthreads();
    for (int off = 1; off < 1024; off <<= 1) {
      const unsigned add = (threadIdx.x >= (unsigned)off) ? buf[threadIdx.x - off] : 0u;
      __syncthreads();
      buf[threadIdx.x] += add;
      __syncthreads();
    }
    if (idx < NBINS) st[idx] = carry + buf[threadIdx.x] - v;
    carry += buf[1023];
    __syncthreads();
  }
}

// voxel ranking key: count desc, vid asc  ==  (count<<17 | (NVOX - vid)) sorted descending
__global__ void k_rkey(const unsigned* __restrict__ counts, unsigned long long* rkeys) {
  const int b = blockIdx.y;
  const int v = blockIdx.x * blockDim.x + threadIdx.x;
  if (v >= VSORT) return;
  unsigned long long k = 0ull;                        // padding sorts last (desc)
  if (v < NVOX) {
    const unsigned c = counts[(size_t)b * CSTRIDE + v];
    k = ((unsigned long long)c << 17) | (unsigned)(NVOX - v);
  }
  rkeys[(size_t)b * VSORT + v] = k;
}

// bitonic compare-exchange, one global pass (stride j >= TILE)
__global__ void k_bitonic_global(unsigned long long* data, int n, int k, int j, int desc) {
  const int b = blockIdx.y;
  data += (size_t)b * n;
  const int t = blockIdx.x * blockDim.x + threadIdx.x;
  if (t >= (n >> 1)) return;
  const int i = ((t & ~(j - 1)) << 1) | (t & (j - 1));
  const int p = i | j;
  bool up = ((i & k) == 0);
  if (desc) up = !up;
  const unsigned long long a = data[i], c = data[p];
  if ((a > c) == up) { data[i] = c; data[p] = a; }
}

// finish strides jstart..1 (jstart <= TILE/2) inside LDS, TILE elems / TTHR thr per block.
// Tile staging uses the CDNA5 async Global<->LDS b128 path when available.
__global__ void k_bitonic_shared(unsigned long long* data, int n, int k, int jstart, int desc) {
  __shared__ unsigned long long lds[TILE];            // 32 KB
  const int b = blockIdx.y;
  data += (size_t)b * n;
  const int base = blockIdx.x * TILE;
#if HAVE_ASYNC_LDS
  {
    gas_v4i* gsrc = to_glob(data + base);             // 16B granules: TILE/2 of them
    las_v4i* ldst = to_lds(lds);
    __builtin_amdgcn_global_load_async_to_lds_b128(gsrc + threadIdx.x, ldst + threadIdx.x, 0, 0);
    __builtin_amdgcn_global_load_async_to_lds_b128(gsrc + threadIdx.x + TTHR,
                                                   ldst + threadIdx.x + TTHR, 0, 0);
    wait_asynccnt0();
  }
  __syncthreads();
#else
  for (int q = 0; q < TILE; q += TTHR)
    lds[threadIdx.x + q] = data[base + threadIdx.x + q];
  __syncthreads();
#endif
  for (int j = jstart; j >= 1; j >>= 1) {
#pragma unroll
    for (int q = 0; q < 2; ++q) {                     // TILE/2 pairs, TTHR threads
      const int t = threadIdx.x + q * TTHR;
      const int i = ((t & ~(j - 1)) << 1) | (t & (j - 1));
      const int p = i | j;
      const int gi = base + i;
      bool up = ((gi & k) == 0);
      if (desc) up = !up;
      const unsigned long long a = lds[i], c = lds[p];
      if ((a > c) == up) { lds[i] = c; lds[p] = a; }
    }
    __syncthreads();
  }
#if HAVE_ASYNC_LDS
  {
    gas_v4i* gdst = to_glob(data + base);
    las_v4i* lsrc = to_lds(lds);
    __builtin_amdgcn_global_store_async_from_lds_b128(gdst + threadIdx.x, lsrc + threadIdx.x, 0, 0);
    __builtin_amdgcn_global_store_async_from_lds_b128(gdst + threadIdx.x + TTHR,
                                                      lsrc + threadIdx.x + TTHR, 0, 0);
    wait_asynccnt0();
  }
#else
  for (int q = 0; q < TILE; q += TTHR)
    data[base + threadIdx.x + q] = lds[threadIdx.x + q];
#endif
}

// gather: one block (64 thr) per top voxel; writes pts and (thread 0) voxel center
__global__ void k_output(const float* __restrict__ pc, const unsigned* __restrict__ baseBits,
                         const unsigned* __restrict__ starts,
                         const unsigned long long* __restrict__ pkeys,
                         const unsigned long long* __restrict__ rkeys,
                         float* __restrict__ out) {
  const int b = blockIdx.y;
  const int v = blockIdx.x;          // 0..VTOP-1, rank in top_k order
  const unsigned t = threadIdx.x;    // 0..63
  const unsigned long long rk = rkeys[(size_t)b * VSORT + v];
  const unsigned cnt = (unsigned)(rk >> 17);
  int vid = NVOX - (int)(rk & 0x1FFFFull);
  if (vid < 0) vid = 0;
  if (vid >= NBINS) vid = NBINS - 1;
  const unsigned c = cnt > 0u ? cnt : 1u;             // jnp.maximum(top_counts, 1)
  unsigned pos = starts[(size_t)b * CSTRIDE + vid] + (t % c);
  if (pos > NPTS - 1) pos = NPTS - 1;                 // JAX OOB gather clamp
  const unsigned long long pk = pkeys[(size_t)b * NPOW2 + pos];
  unsigned pidx = (unsigned)(pk & IDXMASK);
  if (pidx > NPTS - 1) pidx = NPTS - 1;
  const float base = __uint_as_float(baseBits[b]);
  const float* s = pc + ((size_t)b * NPTS + pidx) * 3;
  const size_t o = (((size_t)b * VTOP + v) * PPV + t) * 3;
  out[o + 0] = s[0] / base;
  out[o + 1] = s[1] / base;
  out[o + 2] = s[2] / base;
  if (t == 0) {
    const float jx = (float)(vid / 1600);
    const float jy = (float)((vid / 40) % 40);
    const float jz = (float)(vid % 40);
    const size_t co = (size_t)BATCHES * VTOP * PPV * 3 + ((size_t)b * VTOP + v) * 3;
    out[co + 0] = -1.0f + jx * 0.05f + 0.025f;
    out[co + 1] = -1.0f + jy * 0.05f + 0.025f;
    out[co + 2] = -1.0f + jz * 0.05f + 0.025f;
  }
}

// ---------------- host launcher ----------------
extern "C" void kernel_launch(void* const* d_in, const int* in_sizes, int n_in,
                              void* d_out, int out_size, void* d_ws, size_t ws_size,
                              hipStream_t stream) {
  (void)in_sizes; (void)n_in; (void)out_size; (void)ws_size;
  const float* pc = (const float*)d_in[0];
  float* out = (float*)d_out;
  char* ws = (char*)d_ws;
  unsigned* baseBits = (unsigned*)(ws + WS_BASE);
  unsigned* counts   = (unsigned*)(ws + WS_COUNTS);
  unsigned* starts   = (unsigned*)(ws + WS_STARTS);
  unsigned long long* pkeys = (unsigned long long*)(ws + WS_PKEYS);
  unsigned long long* rkeys = (unsigned long long*)(ws + WS_RKEYS);

  // zero scratch that is accumulated into (every call -> deterministic)
  k_zero<<<dim3(1), dim3(256), 0, stream>>>(baseBits, BATCHES);
  {
    const int n = BATCHES * CSTRIDE;
    k_zero<<<dim3((n + 255) / 256), dim3(256), 0, stream>>>(counts, n);
  }

  k_reduce_base<<<dim3(512, BATCHES), dim3(256), 0, stream>>>(pc, baseBits);
  k_bin<<<dim3(NPOW2 / 256, BATCHES), dim3(256), 0, stream>>>(pc, baseBits, counts, pkeys);
  k_scan<<<dim3(BATCHES), dim3(1024), 0, stream>>>(counts, starts);

  // stable counting sort of points: ascending bitonic over (vid<<19 | idx)
  for (int k = 2; k <= NPOW2; k <<= 1) {
    int j = k >> 1;
    for (; j >= TILE; j >>= 1)
      k_bitonic_global<<<dim3((NPOW2 / 2 + 255) / 256, BATCHES), dim3(256), 0, stream>>>(
          pkeys, NPOW2, k, j, 0);
    k_bitonic_shared<<<dim3(NPOW2 / TILE, BATCHES), dim3(TTHR), 0, stream>>>(
        pkeys, NPOW2, k, j, 0);
  }

  // top_k(counts, 4096): descending bitonic over (count<<17 | (NVOX - vid))
  k_rkey<<<dim3(VSORT / 256, BATCHES), dim3(256), 0, stream>>>(counts, rkeys);
  for (int k = 2; k <= VSORT; k <<= 1) {
    int j = k >> 1;
    for (; j >= TILE; j >>= 1)
      k_bitonic_global<<<dim3((VSORT / 2 + 255) / 256, BATCHES), dim3(256), 0, stream>>>(
          rkeys, VSORT, k, j, 1);
    k_bitonic_shared<<<dim3(VSORT / TILE, BATCHES), dim3(TTHR), 0, stream>>>(
        rkeys, VSORT, k, j, 1);
  }

  k_output<<<dim3(VTOP, BATCHES), dim3(PPV), 0, stream>>>(pc, baseBits, starts, pkeys, rkeys, out);
}